// MHGCN_6184752906287
// MI455X (gfx1250) — compile-verified
//
#include <hip/hip_runtime.h>

#define N_NODES 50000
#define N_REL   3
#define N_EDGES 800000
#define N_FEAT  128
#define N_OUT   64

typedef float v2f __attribute__((ext_vector_type(2)));
typedef float v8f __attribute__((ext_vector_type(8)));

// -------------------------------------------------------------------------
// Y[M x 64] = X[M x K] * W[K x 64], f32 WMMA (V_WMMA_F32_16X16X4_F32).
// M is a multiple of 16 (50000 = 3125*16). One wave per 16x16 output tile;
// 8 waves/block = 2 m-tiles x 4 n-tiles. W staged in LDS (<=32KB).
// -------------------------------------------------------------------------
__global__ void k_gemm_wmma(const float* __restrict__ X, const float* __restrict__ W,
                            float* __restrict__ Y, int M, int K) {
    __shared__ float ldsW[N_FEAT * N_OUT];          // 32 KB max (K=128)
    const int tid = threadIdx.x;
    for (int i = tid; i < K * N_OUT; i += blockDim.x) ldsW[i] = W[i];
    __syncthreads();

    const int lane  = tid & 31;
    const int wave  = tid >> 5;
    const int ntile = wave & 3;                      // 4 tiles across N_OUT=64
    const int mtile = blockIdx.x * 2 + (wave >> 2);  // 2 m-tiles per block
    const int nMTiles = M >> 4;
    if (mtile >= nMTiles) return;                    // wave-uniform exit

    // A-matrix 16x4 f32 layout: lane%16 = row M; lane half selects K pair {0,1}/{2,3}
    const int mrow  = mtile * 16 + (lane & 15);
    const int khalf = (lane >> 4) * 2;               // 0 or 2
    // B-matrix 4x16 layout: lane%16 = col N; lane half selects K rows {0,1}/{2,3}
    const int ncol  = ntile * 16 + (lane & 15);

    const float* xrow = X + (long long)mrow * K;
    v8f c = {};
    for (int k0 = 0; k0 < K; k0 += 4) {
        v2f a, b;
        a.x = xrow[k0 + khalf];
        a.y = xrow[k0 + khalf + 1];
        b.x = ldsW[(k0 + khalf)     * N_OUT + ncol];
        b.y = ldsW[(k0 + khalf + 1) * N_OUT + ncol];
        // 8 args: (neg_a, A, neg_b, B, c_mod, C, reuse_a, reuse_b)
        c = __builtin_amdgcn_wmma_f32_16x16x4_f32(false, a, false, b, (short)0, c,
                                                  false, false);
    }

    // C/D layout: VGPR r -> row (r + 8*(lane/16)), col = lane%16
    const int rbase = mtile * 16 + (lane >> 4) * 8;
    float* yout = Y + (long long)rbase * N_OUT + ntile * 16 + (lane & 15);
#pragma unroll
    for (int r = 0; r < 8; ++r) yout[(long long)r * N_OUT] = c[r];
}

// -------------------------------------------------------------------------
// One wave per edge: gather a 64-float row (one coalesced 256B request as
// float2/lane), scale by wv = vals[e]*weight_b[rel]*scale, scatter with
// hardware f32 atomics. L2-resident (12.8MB accumulator << 192MB L2).
// -------------------------------------------------------------------------
__global__ void k_scatter(const float* __restrict__ sup, const int* __restrict__ src,
                          const int* __restrict__ dst, const float* __restrict__ vals,
                          const float* __restrict__ wb, float* __restrict__ acc,
                          float scale) {
    const long long t = (long long)blockIdx.x * blockDim.x + threadIdx.x;
    const long long e = t >> 5;
    const int lane    = (int)(t & 31);
    if (e >= (long long)N_REL * N_EDGES) return;

    const int   rel = (int)(e / N_EDGES);
    const float w   = vals[e] * wb[rel] * scale;
    const int   sN  = src[e];
    const int   dN  = dst[e];

    const float2 v = ((const float2*)(sup + (long long)sN * N_OUT))[lane];
    float* drow = acc + (long long)dN * N_OUT + lane * 2;
    unsafeAtomicAdd(drow,     w * v.x);   // global_atomic_add_f32, no CAS loop
    unsafeAtomicAdd(drow + 1, w * v.y);
}

__global__ void k_zero(float* __restrict__ p, long long n) {
    long long i = (long long)blockIdx.x * blockDim.x + threadIdx.x;
    if (i < n) p[i] = 0.0f;
}

__global__ void k_bias(float* __restrict__ p, const float* __restrict__ b, long long n) {
    long long i = (long long)blockIdx.x * blockDim.x + threadIdx.x;
    if (i < n) p[i] += b[i & (N_OUT - 1)];
}

// out = 0.5 * (U1 + b2); scatter2 then adds 0.5*wv*support2 rows -> (U1+U2)/2
__global__ void k_init_out(float* __restrict__ out, const float* __restrict__ U1,
                           const float* __restrict__ b2, long long n) {
    long long i = (long long)blockIdx.x * blockDim.x + threadIdx.x;
    if (i < n) out[i] = 0.5f * (U1[i] + b2[i & (N_OUT - 1)]);
}

extern "C" void kernel_launch(void* const* d_in, const int* in_sizes, int n_in,
                              void* d_out, int out_size, void* d_ws, size_t ws_size,
                              hipStream_t stream) {
    const float* feature = (const float*)d_in[0];
    const int*   src     = (const int*)  d_in[1];
    const int*   dst     = (const int*)  d_in[2];
    const float* vals    = (const float*)d_in[3];
    const float* wb      = (const float*)d_in[4];
    const float* W1      = (const float*)d_in[5];
    const float* b1      = (const float*)d_in[6];
    const float* W2      = (const float*)d_in[7];
    const float* b2      = (const float*)d_in[8];
    float*       out     = (float*)d_out;

    float* bufA = (float*)d_ws;                       // support (12.8 MB)
    float* bufB = bufA + (size_t)N_NODES * N_OUT;     // agg1/U1  (12.8 MB)

    const long long nElems = (long long)N_NODES * N_OUT;      // 3.2M
    const long long nEdges = (long long)N_REL * N_EDGES;      // 2.4M
    const int tpb = 256;
    const int elemBlocks = (int)((nElems + tpb - 1) / tpb);
    const long long scatterThreads = nEdges * 32;
    const int scatterBlocks = (int)((scatterThreads + tpb - 1) / tpb);
    const int gemmBlocks = (N_NODES / 16 + 1) / 2;            // 2 m-tiles/block

    // Layer 1
    k_gemm_wmma<<<gemmBlocks, 256, 0, stream>>>(feature, W1, bufA, N_NODES, N_FEAT);
    k_zero<<<elemBlocks, tpb, 0, stream>>>(bufB, nElems);
    k_scatter<<<scatterBlocks, tpb, 0, stream>>>(bufA, src, dst, vals, wb, bufB, 1.0f);
    k_bias<<<elemBlocks, tpb, 0, stream>>>(bufB, b1, nElems);     // bufB = U1

    // Layer 2 (fold /2 and b2 into init + scatter scale)
    k_gemm_wmma<<<gemmBlocks, 256, 0, stream>>>(bufB, W2, bufA, N_NODES, N_OUT);
    k_init_out<<<elemBlocks, tpb, 0, stream>>>(out, bufB, b2, nElems);
    k_scatter<<<scatterBlocks, tpb, 0, stream>>>(bufA, src, dst, vals, wb, out, 0.5f);
}